// Heterogeneous_Attention_Layer_18262200943352
// MI455X (gfx1250) — compile-verified
//
#include <hip/hip_runtime.h>
#include <hip/hip_bf16.h>
#include <float.h>

#define N_ITEM 150000
#define N_USER 150000
#define D 64
#define H 4
#define HD 16
#define E_BUYS 750000
#define E_REL 750000
#define EPS 1e-5f

typedef __attribute__((ext_vector_type(2))) float v2f;
typedef __attribute__((ext_vector_type(8))) float v8f;

// ---------------------------------------------------------------- fill
__global__ void fillf(float* p, size_t n, float v) {
    size_t i = (size_t)blockIdx.x * blockDim.x + threadIdx.x;
    size_t stride = (size_t)gridDim.x * blockDim.x;
    for (; i < n; i += stride) p[i] = v;
}

// ---------------------------------------------------------------- batchnorm stats
__global__ void bn_stats(const float* __restrict__ x, int n,
                         float* __restrict__ sum, float* __restrict__ sq) {
    int gtid = blockIdx.x * blockDim.x + threadIdx.x;
    int d = gtid & 63;
    int r = gtid >> 6;
    int rstride = (gridDim.x * blockDim.x) >> 6;
    float s = 0.f, s2 = 0.f;
    for (; r < n; r += rstride) {
        float v = x[(size_t)r * D + d];
        s += v; s2 += v * v;
    }
    atomicAdd(&sum[d], s);
    atomicAdd(&sq[d], s2);
}

__global__ void bn_finalize(const float* __restrict__ sum, const float* __restrict__ sq,
                            const float* __restrict__ g, const float* __restrict__ b,
                            float n, float* __restrict__ scale, float* __restrict__ shift) {
    int d = threadIdx.x;
    if (d < D) {
        float mu = sum[d] / n;
        float var = sq[d] / n - mu * mu;
        float sc = g[d] * rsqrtf(var + EPS);
        scale[d] = sc;
        shift[d] = b[d] - mu * sc;
    }
}

// ---------------------------------------------------------------- multi-output WMMA GEMM
// One wave32 per 16-row tile. out = bn(x) @ W (+ bias). K=D=64 via 16 steps of
// V_WMMA_F32_16X16X4_F32 (fp32 precision, matches fp32 reference).
__global__ __launch_bounds__(256)
void gemm_multi(const float* __restrict__ x, int nrows,
                const float* __restrict__ scale, const float* __restrict__ shift,
                const float* W0, const float* B0, float* O0,
                const float* W1, const float* B1, float* O1,
                const float* W2, const float* B2, float* O2,
                const float* W3, const float* B3, float* O3,
                int nout) {
    const float* Ws[4] = {W0, W1, W2, W3};
    const float* Bs[4] = {B0, B1, B2, B3};
    float*       Os[4] = {O0, O1, O2, O3};

    int wid  = (blockIdx.x * blockDim.x + threadIdx.x) >> 5;
    int lane = threadIdx.x & 31;
    int row0 = wid * 16;
    if (row0 >= nrows) return;            // wave-uniform: EXEC stays all-ones

    int m    = lane & 15;
    int lh   = lane >> 4;                  // which half of the wave
    int koff = lh * 2;

    const float* xr = x + (size_t)(row0 + m) * D;
    v2f a[16];
    #pragma unroll
    for (int ks = 0; ks < 16; ++ks) {
        int k = ks * 4 + koff;
        a[ks].x = xr[k]     * scale[k]     + shift[k];
        a[ks].y = xr[k + 1] * scale[k + 1] + shift[k + 1];
    }

    for (int o = 0; o < nout; ++o) {
        const float* W = Ws[o];
        const float* B = Bs[o];
        float*       O = Os[o];
        #pragma unroll
        for (int nt = 0; nt < 4; ++nt) {
            int n = nt * 16 + m;
            float bv = B ? B[n] : 0.0f;
            v8f acc;
            #pragma unroll
            for (int r = 0; r < 8; ++r) acc[r] = bv;
            #pragma unroll
            for (int ks = 0; ks < 16; ++ks) {
                int k = ks * 4 + koff;
                v2f bf;
                bf.x = W[(size_t)k * D + n];
                bf.y = W[(size_t)(k + 1) * D + n];
                acc = __builtin_amdgcn_wmma_f32_16x16x4_f32(
                          false, a[ks], false, bf, (short)0, acc, false, false);
            }
            int mb = row0 + lh * 8;
            #pragma unroll
            for (int r = 0; r < 8; ++r) O[(size_t)(mb + r) * D + n] = acc[r];
        }
    }
}

// ---------------------------------------------------------------- edge pass 1: score + segment max
// 64 threads per edge (one per feature), two waves; per-head 16-lane shuffle reduction.
__global__ __launch_bounds__(256)
void edge_score(const float* __restrict__ q, const float* __restrict__ k,
                const float* __restrict__ att,
                const int* __restrict__ src, const int* __restrict__ dst,
                const int* __restrict__ cnt, const float* __restrict__ emb,
                int ne, float* __restrict__ ebuf, float* __restrict__ mbuf) {
    long long t = (long long)blockIdx.x * blockDim.x + threadIdx.x;
    int e = (int)(t >> 6);
    if (e >= ne) return;
    int d = (int)(t & 63);
    int s  = src[e];
    int dd = dst[e];
    float val = q[(size_t)s * D + d] + k[(size_t)dd * D + d];
    if (cnt) val += emb[(size_t)cnt[e] * D + d];
    float sg = att[d] / (1.0f + __expf(-val));
    // reduce 16 lanes per head (heads align with 16-lane groups inside a wave32)
    #pragma unroll
    for (int off = 8; off >= 1; off >>= 1) sg += __shfl_xor(sg, off, 16);
    if ((d & 15) == 0) {
        int h = d >> 4;
        ebuf[(size_t)e * H + h] = sg;
        atomicMax(&mbuf[(size_t)dd * H + h], sg);
    }
}

// ---------------------------------------------------------------- edge pass 2: exp + denominator
__global__ __launch_bounds__(256)
void edge_exp(const float* __restrict__ mbuf, const int* __restrict__ dst,
              int ne, float* __restrict__ ebuf, float* __restrict__ den) {
    long long t = (long long)blockIdx.x * blockDim.x + threadIdx.x;
    int e = (int)(t >> 2);
    if (e >= ne) return;
    int h = (int)(t & 3);
    int dd = dst[e];
    float ex = __expf(ebuf[(size_t)e * H + h] - mbuf[(size_t)dd * H + h]);
    ebuf[(size_t)e * H + h] = ex;
    atomicAdd(&den[(size_t)dd * H + h], ex);
}

// ---------------------------------------------------------------- edge pass 3: weighted scatter
__global__ __launch_bounds__(256)
void edge_agg(const float* __restrict__ v, const float* __restrict__ ebuf,
              const float* __restrict__ den,
              const int* __restrict__ src, const int* __restrict__ dst,
              int ne, float* __restrict__ agg) {
    long long t = (long long)blockIdx.x * blockDim.x + threadIdx.x;
    int e = (int)(t >> 6);
    if (e >= ne) return;
    int d = (int)(t & 63);
    int h = d >> 4;
    int s  = src[e];
    int dd = dst[e];
    float a = ebuf[(size_t)e * H + h] / den[(size_t)dd * H + h];
    atomicAdd(&agg[(size_t)dd * D + d], v[(size_t)s * D + d] * a);
}

// ---------------------------------------------------------------- final: relu(agg@W_agg + b_agg + bn(fi)@W_self)
__global__ __launch_bounds__(256)
void final_out(const float* __restrict__ agg, const float* __restrict__ fi,
               const float* __restrict__ scale, const float* __restrict__ shift,
               const float* __restrict__ Wagg, const float* __restrict__ bagg,
               const float* __restrict__ Wself, float* __restrict__ out, int nrows) {
    int wid  = (blockIdx.x * blockDim.x + threadIdx.x) >> 5;
    int lane = threadIdx.x & 31;
    int row0 = wid * 16;
    if (row0 >= nrows) return;

    int m    = lane & 15;
    int lh   = lane >> 4;
    int koff = lh * 2;

    const float* ar = agg + (size_t)(row0 + m) * D;
    const float* xr = fi  + (size_t)(row0 + m) * D;
    v2f a1[16], a2[16];
    #pragma unroll
    for (int ks = 0; ks < 16; ++ks) {
        int k = ks * 4 + koff;
        a1[ks].x = ar[k];
        a1[ks].y = ar[k + 1];
        a2[ks].x = xr[k]     * scale[k]     + shift[k];
        a2[ks].y = xr[k + 1] * scale[k + 1] + shift[k + 1];
    }
    #pragma unroll
    for (int nt = 0; nt < 4; ++nt) {
        int n = nt * 16 + m;
        float bv = bagg[n];
        v8f acc;
        #pragma unroll
        for (int r = 0; r < 8; ++r) acc[r] = bv;
        #pragma unroll
        for (int ks = 0; ks < 16; ++ks) {
            int k = ks * 4 + koff;
            v2f bf;
            bf.x = Wagg[(size_t)k * D + n];
            bf.y = Wagg[(size_t)(k + 1) * D + n];
            acc = __builtin_amdgcn_wmma_f32_16x16x4_f32(
                      false, a1[ks], false, bf, (short)0, acc, false, false);
        }
        #pragma unroll
        for (int ks = 0; ks < 16; ++ks) {
            int k = ks * 4 + koff;
            v2f bf;
            bf.x = Wself[(size_t)k * D + n];
            bf.y = Wself[(size_t)(k + 1) * D + n];
            acc = __builtin_amdgcn_wmma_f32_16x16x4_f32(
                      false, a2[ks], false, bf, (short)0, acc, false, false);
        }
        int mb = row0 + lh * 8;
        #pragma unroll
        for (int r = 0; r < 8; ++r)
            out[(size_t)(mb + r) * D + n] = fmaxf(acc[r], 0.0f);
    }
}

// ---------------------------------------------------------------- host launch
extern "C" void kernel_launch(void* const* d_in, const int* in_sizes, int n_in,
                              void* d_out, int out_size, void* d_ws, size_t ws_size,
                              hipStream_t stream) {
    const float* ft_item   = (const float*)d_in[0];
    const float* ft_user   = (const float*)d_in[1];
    const float* bn_g_item = (const float*)d_in[2];
    const float* bn_b_item = (const float*)d_in[3];
    const float* bn_g_user = (const float*)d_in[4];
    const float* bn_b_user = (const float*)d_in[5];
    const float* Wq_b = (const float*)d_in[6];
    const float* bq_b = (const float*)d_in[7];
    const float* Wk_b = (const float*)d_in[8];
    const float* Wv_b = (const float*)d_in[9];
    const float* att_b = (const float*)d_in[10];
    const float* emb_cnt = (const float*)d_in[11];
    const float* Wq_r = (const float*)d_in[12];
    const float* bq_r = (const float*)d_in[13];
    const float* Wk_r = (const float*)d_in[14];
    const float* Wv_r = (const float*)d_in[15];
    const float* att_r = (const float*)d_in[16];
    const float* W_agg = (const float*)d_in[17];
    const float* b_agg = (const float*)d_in[18];
    const float* W_self = (const float*)d_in[19];
    const int* buys_src = (const int*)d_in[20];
    const int* buys_dst = (const int*)d_in[21];
    const int* buys_cnt = (const int*)d_in[22];
    const int* rel_src  = (const int*)d_in[23];
    const int* rel_dst  = (const int*)d_in[24];
    float* out = (float*)d_out;

    // workspace layout (floats)
    float* ws = (float*)d_ws;
    size_t o = 0;
    float* sum_i = ws + o; o += 64;
    float* sq_i  = ws + o; o += 64;
    float* sum_u = ws + o; o += 64;
    float* sq_u  = ws + o; o += 64;
    float* sc_i  = ws + o; o += 64;
    float* sh_i  = ws + o; o += 64;
    float* sc_u  = ws + o; o += 64;
    float* sh_u  = ws + o; o += 64;
    float* qb  = ws + o; o += (size_t)N_USER * D;
    float* vb  = ws + o; o += (size_t)N_USER * D;
    float* kb  = ws + o; o += (size_t)N_ITEM * D;
    float* qr  = ws + o; o += (size_t)N_ITEM * D;
    float* kr  = ws + o; o += (size_t)N_ITEM * D;
    float* vr  = ws + o; o += (size_t)N_ITEM * D;
    float* agg = ws + o; o += (size_t)N_ITEM * D;
    float* mbuf = ws + o; o += (size_t)N_ITEM * H;
    float* den  = ws + o; o += (size_t)N_ITEM * H;
    float* ebuf = ws + o; o += (size_t)(E_BUYS + E_REL) * H;

    const int TPB = 256;
    auto blocks_for = [](size_t n) {
        size_t b = (n + 255) / 256;
        return (int)(b > 16384 ? 16384 : b);
    };

    // init
    fillf<<<1, TPB, 0, stream>>>(sum_i, 512, 0.0f);  // all stats+scale/shift
    fillf<<<blocks_for((size_t)N_ITEM * D), TPB, 0, stream>>>(agg, (size_t)N_ITEM * D, 0.0f);
    fillf<<<blocks_for((size_t)N_ITEM * H), TPB, 0, stream>>>(den, (size_t)N_ITEM * H, 0.0f);
    fillf<<<blocks_for((size_t)N_ITEM * H), TPB, 0, stream>>>(mbuf, (size_t)N_ITEM * H, -FLT_MAX);

    // batchnorm
    bn_stats<<<1024, TPB, 0, stream>>>(ft_item, N_ITEM, sum_i, sq_i);
    bn_stats<<<1024, TPB, 0, stream>>>(ft_user, N_USER, sum_u, sq_u);
    bn_finalize<<<1, 64, 0, stream>>>(sum_i, sq_i, bn_g_item, bn_b_item, (float)N_ITEM, sc_i, sh_i);
    bn_finalize<<<1, 64, 0, stream>>>(sum_u, sq_u, bn_g_user, bn_b_user, (float)N_USER, sc_u, sh_u);

    // q/k/v projections (WMMA): user side (q_b, v_b), item side (k_b, q_r, k_r, v_r)
    int tilesU = (N_USER + 15) / 16;
    int tilesI = (N_ITEM + 15) / 16;
    int gU = (tilesU + 7) / 8;   // 8 waves per 256-thread block
    int gI = (tilesI + 7) / 8;
    gemm_multi<<<gU, TPB, 0, stream>>>(ft_user, N_USER, sc_u, sh_u,
                                       Wq_b, bq_b, qb,
                                       Wv_b, nullptr, vb,
                                       nullptr, nullptr, nullptr,
                                       nullptr, nullptr, nullptr, 2);
    gemm_multi<<<gI, TPB, 0, stream>>>(ft_item, N_ITEM, sc_i, sh_i,
                                       Wk_b, nullptr, kb,
                                       Wq_r, bq_r, qr,
                                       Wk_r, nullptr, kr,
                                       Wv_r, nullptr, vr, 4);

    // edge softmax (3 passes), buys then rel, shared segment buffers over items
    long long tb = (long long)E_BUYS * 64, tr = (long long)E_REL * 64;
    edge_score<<<(int)((tb + TPB - 1) / TPB), TPB, 0, stream>>>(
        qb, kb, att_b, buys_src, buys_dst, buys_cnt, emb_cnt, E_BUYS, ebuf, mbuf);
    edge_score<<<(int)((tr + TPB - 1) / TPB), TPB, 0, stream>>>(
        qr, kr, att_r, rel_src, rel_dst, nullptr, nullptr, E_REL, ebuf + (size_t)E_BUYS * H, mbuf);

    long long eb4 = (long long)E_BUYS * H, er4 = (long long)E_REL * H;
    edge_exp<<<(int)((eb4 + TPB - 1) / TPB), TPB, 0, stream>>>(mbuf, buys_dst, E_BUYS, ebuf, den);
    edge_exp<<<(int)((er4 + TPB - 1) / TPB), TPB, 0, stream>>>(mbuf, rel_dst, E_REL, ebuf + (size_t)E_BUYS * H, den);

    edge_agg<<<(int)((tb + TPB - 1) / TPB), TPB, 0, stream>>>(
        vb, ebuf, den, buys_src, buys_dst, E_BUYS, agg);
    edge_agg<<<(int)((tr + TPB - 1) / TPB), TPB, 0, stream>>>(
        vr, ebuf + (size_t)E_BUYS * H, den, rel_src, rel_dst, E_REL, agg);

    // output
    final_out<<<gI, TPB, 0, stream>>>(agg, ft_item, sc_i, sh_i, W_agg, b_agg, W_self, out, N_ITEM);
}